// ResidualQuantizer_19396072309111
// MI455X (gfx1250) — compile-verified
//
#include <hip/hip_runtime.h>
#include <hip/hip_bf16.h>

typedef float v2f __attribute__((ext_vector_type(2)));
typedef float v4f __attribute__((ext_vector_type(4)));
typedef float v8f __attribute__((ext_vector_type(8)));

// Problem geometry (fixed by the reference)
#define NB        8
#define DDIM      256
#define HW        1024
#define NROWS     8192      // NB*HW residual rows
#define NE        16384     // codebook entries
#define SEGS      4
#define EPSEG     (NE / SEGS)       // 4096 entries per segment
#define CHUNK     64                // entries staged in LDS at a time
#define NCHUNK    (EPSEG / CHUNK)   // 64 chunks per segment
#define LDS_ROW   260               // 256 dwords + 4 pad dwords (bank-conflict-free b64 reads)
#define BUF_DW    (CHUNK * LDS_ROW) // dwords per LDS buffer
#define ROWS_PB   128               // rows per block (8 waves x 16 rows)

// Output layout (flat, return order: z_hat_out, loss, total_idx)
#define ZH_OFF    0
#define LOSS_OFF  2097152
#define IDX_OFF   2097153

// Workspace layout (in floats)
#define WS_HN      0                       // 16384 floats: half squared norms
#define WS_PBEST   16384                   // SEGS*NROWS floats
#define WS_PIDX    (16384 + SEGS*NROWS)    // SEGS*NROWS ints
#define WS_PART    (16384 + 2*SEGS*NROWS)  // 256 floats: loss partials

// ---------------------------------------------------------------------------
// Kernel A: half squared norms of codebook rows. One wave32 per row.
// ---------------------------------------------------------------------------
__global__ __launch_bounds__(256) void k_norms(const float* __restrict__ emb,
                                               float* __restrict__ hn) {
    const int wave = threadIdx.x >> 5;
    const int lane = threadIdx.x & 31;
    const int r = blockIdx.x * 8 + wave;
    const float* row = emb + (size_t)r * DDIM;
    float s = 0.f;
#pragma unroll
    for (int j = 0; j < 8; ++j) {
        float v = row[lane + 32 * j];
        s += v * v;
    }
#pragma unroll
    for (int m = 16; m >= 1; m >>= 1) s += __shfl_xor(s, m, 32);
    if (lane == 0) hn[r] = 0.5f * s;
}

// ---------------------------------------------------------------------------
// Async staging: 64 entries x 256 floats streamed memory->LDS with
// GLOBAL_LOAD_ASYNC_TO_LDS_B128 (ASYNCcnt-tracked, no data VGPRs).
// GVS addressing: uniform SGPR base (emb) + per-lane 32-bit byte offset.
// LDS address = low 32 bits of the generic pointer (flat->LDS truncation).
// ---------------------------------------------------------------------------
__device__ __forceinline__ void stage_chunk_async(const float* __restrict__ emb,
                                                  float* ldsBuf, int eBase0, int t) {
#pragma unroll
    for (int i = 0; i < 16; ++i) {
        const int linear = t + 256 * i;          // 0..4095 float4s
        const int e  = linear >> 6;
        const int d4 = linear & 63;
        const uint32_t ldsByte = (uint32_t)(uintptr_t)&ldsBuf[e * LDS_ROW + d4 * 4];
        const uint32_t voff = (uint32_t)((eBase0 + e) * (DDIM * 4) + d4 * 16);
        asm volatile("global_load_async_to_lds_b128 %0, %1, %2"
                     :: "v"(ldsByte), "v"(voff), "s"(emb)
                     : "memory");
    }
}

// ---------------------------------------------------------------------------
// Kernel B: fused GEMM + running argmax via V_WMMA_F32_16X16X4_F32 with a
// double-buffered async-to-LDS codebook pipeline.
// grid = (64 row-blocks, 4 entry segments), block = 256 threads (8 waves).
// Each wave owns 16 residual rows; the A stripe lives in registers as WMMA
// fragments and is reused against every staged codebook tile.
// ---------------------------------------------------------------------------
__global__ __launch_bounds__(256) void k_argmax(const float* __restrict__ z,
                                                const float* __restrict__ emb,
                                                const float* __restrict__ hn,
                                                float* __restrict__ pbest,
                                                int* __restrict__ pidx) {
    __shared__ float ldsE[2 * BUF_DW];        // 133,120 B (double buffer)

    const int t    = threadIdx.x;
    const int wave = t >> 5;
    const int lane = t & 31;
    const int m    = lane & 15;               // row-in-stripe (A) / entry-col (B,C)
    const int hi   = lane >> 4;               // lane half
    const int klo  = hi ? 2 : 0;              // K sub-offset per ISA A/B layout

    const int rowBase = blockIdx.x * ROWS_PB + wave * 16;
    const int segBase = blockIdx.y * EPSEG;

    // Kick off chunk 0 before doing anything else.
    stage_chunk_async(emb, &ldsE[0], segBase, t);

    // ---- Preload A fragments: row n = rowBase + m, residual[n][d] = z[b][d][hw]
    const int n  = rowBase + m;
    const int b  = n >> 10;
    const int hw = n & 1023;
    const float* zrow = z + (size_t)b * (DDIM * HW) + hw;
    v2f aReg[64];
#pragma unroll
    for (int ks = 0; ks < 64; ++ks) {
        const int d0 = ks * 4 + klo;
        aReg[ks].x = zrow[(size_t)d0 * HW];
        aReg[ks].y = zrow[(size_t)(d0 + 1) * HW];
    }

    float vmax[8];
    int   vidx[8];
#pragma unroll
    for (int v = 0; v < 8; ++v) { vmax[v] = -3.402823466e38f; vidx[v] = 0x7fffffff; }

    for (int chunk = 0; chunk < NCHUNK; ++chunk) {
        const int eBase0 = segBase + chunk * CHUNK;
        float* cur = &ldsE[(chunk & 1) * BUF_DW];

        if (chunk + 1 < NCHUNK) {
            // Prefetch the next chunk into the other buffer, then wait for the
            // 16 in-flight asyncs of the CURRENT chunk (in-order completion),
            // leaving the 16 just-issued ones pending.
            stage_chunk_async(emb, &ldsE[((chunk + 1) & 1) * BUF_DW],
                              eBase0 + CHUNK, t);
            asm volatile("s_wait_asynccnt 0x10" ::: "memory");
        } else {
            asm volatile("s_wait_asynccnt 0x0" ::: "memory");
        }
        __syncthreads();   // all waves' current-chunk data visible in LDS

        for (int sub = 0; sub < 4; ++sub) {
            const int eCol = eBase0 + sub * 16 + m;      // this lane's entry column
            const float h = hn[eCol];
            const int ldsOff = (sub * 16 + m) * LDS_ROW + klo;

            v8f c = {0.f, 0.f, 0.f, 0.f, 0.f, 0.f, 0.f, 0.f};
#pragma unroll
            for (int ks = 0; ks < 64; ++ks) {
                v2f bf = *(const v2f*)&cur[ldsOff + ks * 4];   // ds_load_2addr_b64
                c = __builtin_amdgcn_wmma_f32_16x16x4_f32(
                        /*neg_a=*/false, aReg[ks],
                        /*neg_b=*/false, bf,
                        /*c_mod=*/(short)0, c,
                        /*reuse_a=*/false, /*reuse_b=*/false);
            }
            // Running argmax with argmin-of-distance tiebreak (lowest index).
#pragma unroll
            for (int v = 0; v < 8; ++v) {
                const float s = c[v] - h;
                if (s > vmax[v] || (s == vmax[v] && eCol < vidx[v])) {
                    vmax[v] = s;
                    vidx[v] = eCol;
                }
            }
        }
        __syncthreads();   // everyone done reading `cur` before it is rewritten
    }

    // C layout: vgpr v -> row v (lanes 0-15) / row v+8 (lanes 16-31).
    // Reduce the 16 entry-columns within each lane half.
#pragma unroll
    for (int v = 0; v < 8; ++v) {
        float s = vmax[v];
        int   id = vidx[v];
#pragma unroll
        for (int msk = 8; msk >= 1; msk >>= 1) {
            const float so = __shfl_xor(s, msk, 32);
            const int   io = __shfl_xor(id, msk, 32);
            if (so > s || (so == s && io < id)) { s = so; id = io; }
        }
        if (m == 0) {
            const int row = rowBase + v + (hi ? 8 : 0);
            pbest[blockIdx.y * NROWS + row] = s;
            pidx [blockIdx.y * NROWS + row] = id;
        }
    }
}

// ---------------------------------------------------------------------------
// Kernel C: merge segment winners, gather codebook, write z_hat_out and
// tiled idx, accumulate deterministic per-block loss partials.
// block = 256 threads = 8 d-groups x 32 rows; grid = 256.
// ---------------------------------------------------------------------------
__global__ __launch_bounds__(256) void k_out(const float* __restrict__ z,
                                             const float* __restrict__ emb,
                                             const float* __restrict__ pbest,
                                             const int* __restrict__ pidx,
                                             float* __restrict__ out,
                                             float* __restrict__ partial) {
    __shared__ int   eSh[32];
    __shared__ float red[256];

    const int t  = threadIdx.x;
    const int tn = t & 31;      // row within block's 32-row slab
    const int td = t >> 5;      // d-group 0..7
    const int n  = blockIdx.x * 32 + tn;

    if (td == 0) {
        float bs = pbest[n];
        int   bi = pidx[n];
#pragma unroll
        for (int s = 1; s < SEGS; ++s) {
            const float s2 = pbest[s * NROWS + n];
            const int   i2 = pidx [s * NROWS + n];
            if (s2 > bs || (s2 == bs && i2 < bi)) { bs = s2; bi = i2; }
        }
        eSh[tn] = bi;
    }
    __syncthreads();

    const int e  = eSh[tn];
    const int b  = n >> 10;
    const int hw = n & 1023;
    const size_t zbase = (size_t)b * (DDIM * HW) + hw;
    const float* erow = emb + (size_t)e * DDIM;

    float acc = 0.f;
#pragma unroll
    for (int j = 0; j < 32; ++j) {
        const int d = td + j * 8;
        const float q  = erow[d];
        const float zz = z[zbase + (size_t)d * HW];
        const float zh = 4.f * q;                       // z_hat after 4 scales
        out[ZH_OFF + zbase + (size_t)d * HW] = zz + (zh - zz);  // straight-through
        // sum_{k=1..4} (k*q - zz)^2 = 30 q^2 - 20 q zz + 4 zz^2
        acc += 30.f * q * q - 20.f * q * zz + 4.f * zz * zz;
    }

    // total_idx = idx tiled SCALE_NUM times along axis 1: [8][4*32][32]
    if (td < 4) out[IDX_OFF + (size_t)b * 4096 + td * 1024 + hw] = (float)e;

    red[t] = acc;
    __syncthreads();
    for (int s = 128; s > 0; s >>= 1) {
        if (t < s) red[t] += red[t + s];
        __syncthreads();
    }
    if (t == 0) partial[blockIdx.x] = red[0];
}

// ---------------------------------------------------------------------------
// Kernel D: deterministic final loss reduction. 1 block, 256 threads.
// loss = (1+BETA)/SCALE_NUM * (sum / numel), BETA=0.25, numel=2097152
// ---------------------------------------------------------------------------
__global__ __launch_bounds__(256) void k_loss(const float* __restrict__ partial,
                                              float* __restrict__ out) {
    __shared__ float red[256];
    const int t = threadIdx.x;
    red[t] = partial[t];
    __syncthreads();
    for (int s = 128; s > 0; s >>= 1) {
        if (t < s) red[t] += red[t + s];
        __syncthreads();
    }
    if (t == 0) out[LOSS_OFF] = red[0] * (1.25f / (4.f * 2097152.f));
}

// ---------------------------------------------------------------------------
extern "C" void kernel_launch(void* const* d_in, const int* in_sizes, int n_in,
                              void* d_out, int out_size, void* d_ws, size_t ws_size,
                              hipStream_t stream) {
    (void)in_sizes; (void)n_in; (void)out_size; (void)ws_size;
    const float* z   = (const float*)d_in[0];   // [8,256,32,32]
    const float* emb = (const float*)d_in[1];   // [16384,256]
    float* out = (float*)d_out;

    float* ws      = (float*)d_ws;
    float* hn      = ws + WS_HN;
    float* pbest   = ws + WS_PBEST;
    int*   pidx    = (int*)(ws + WS_PIDX);
    float* partial = ws + WS_PART;

    k_norms<<<dim3(NE / 8), dim3(256), 0, stream>>>(emb, hn);
    k_argmax<<<dim3(NROWS / ROWS_PB, SEGS), dim3(256), 0, stream>>>(z, emb, hn, pbest, pidx);
    k_out<<<dim3(NROWS / 32), dim3(256), 0, stream>>>(z, emb, pbest, pidx, out, partial);
    k_loss<<<dim3(1), dim3(256), 0, stream>>>(partial, out);
}